// GraphSAGE_26104811225563
// MI455X (gfx1250) — compile-verified
//
#include <hip/hip_runtime.h>

// ---------------------------------------------------------------------------
// GraphSAGE (3-layer, mean aggr) for MI455X / gfx1250, wave32.
//   agg[dst] += x[src]  (fp32 atomics, 512B per edge-row, b128 loads)
//   h = PReLU( (agg*inv_deg) @ Wl^T + x @ Wr^T + b )   via V_WMMA_F32_16X16X4_F32
// Degree is edge-structure-only -> computed once, reused by all 3 layers.
// ---------------------------------------------------------------------------

typedef __attribute__((ext_vector_type(2))) float v2f;
typedef __attribute__((ext_vector_type(8))) float v8f;

#define N_NODES 50000
#define N_EDGES 800000
#define DIM_IN  128
#define DIM_HID 128
#define DIM_OUT 64

// ---------------- degree / inverse count ----------------
__global__ void degree_kernel(const int* __restrict__ dst, float* __restrict__ cnt, int E) {
    int e = blockIdx.x * blockDim.x + threadIdx.x;
    if (e < E) atomicAdd(&cnt[dst[e]], 1.0f);
}

__global__ void invcnt_kernel(float* __restrict__ cnt, int n) {
    int i = blockIdx.x * blockDim.x + threadIdx.x;
    if (i < n) cnt[i] = 1.0f / fmaxf(cnt[i], 1.0f);
}

// ---------------- edge scatter: one wave per edge, 4 floats per lane --------
__global__ void scatter_kernel(const float* __restrict__ feat,
                               const int* __restrict__ src,
                               const int* __restrict__ dst,
                               float* __restrict__ agg, int E) {
    unsigned gid  = blockIdx.x * blockDim.x + threadIdx.x;
    unsigned e    = gid >> 5;           // wave-uniform edge id
    if (e >= (unsigned)E) return;
    int lane = gid & 31;
    int s = src[e];
    int d = dst[e];
    const float4 v = *reinterpret_cast<const float4*>(feat + (size_t)s * DIM_IN + lane * 4);
    float* base = agg + (size_t)d * DIM_IN + lane * 4;
    atomicAdd(base + 0, v.x);
    atomicAdd(base + 1, v.y);
    atomicAdd(base + 2, v.z);
    atomicAdd(base + 3, v.w);
}

// ---------------- fused SAGE GEMM: one wave per 16x16 D tile ----------------
// out[m,n] = sum_k ( agg[m,k]*inv[m] * wl[n,k]  +  xin[m,k] * wr[n,k] ) + b[n]
// A 16x4 f32 frag: lanes 0-15 row M=lane, K={kb,kb+1}; lanes 16-31 K={kb+2,kb+3}
// B 4x16 f32 frag: symmetric (row K striped across lanes).
// C/D 16x16 f32: 8 VGPRs; VGPR r holds M=r (lanes 0-15) / M=r+8 (lanes 16-31).
template <int OUT_DIM, bool PRELU>
__global__ void sage_gemm_kernel(const float* __restrict__ agg,
                                 const float* __restrict__ inv,
                                 const float* __restrict__ xin,
                                 const float* __restrict__ wl,
                                 const float* __restrict__ wr,
                                 const float* __restrict__ bias,
                                 const float* __restrict__ slope,
                                 float* __restrict__ out,
                                 int totalTiles) {
    constexpr int IN = 128;
    constexpr int colTiles = OUT_DIM / 16;
    int wave = (int)((blockIdx.x * blockDim.x + threadIdx.x) >> 5);
    if (wave >= totalTiles) return;          // wave-granular: EXEC stays all-1s
    int lane  = threadIdx.x & 31;
    int tr    = wave / colTiles;
    int tc    = wave % colTiles;
    int l15   = lane & 15;
    int hi    = lane >> 4;                   // 0 for lanes 0-15, 1 for 16-31
    int m     = tr * 16 + l15;
    int n     = tc * 16 + l15;
    int khalf = hi * 2;

    const float  invm   = inv[m];
    const float* aggRow = agg + (size_t)m * IN;
    const float* xRow   = xin + (size_t)m * IN;
    const float* wlRow  = wl  + (size_t)n * IN;
    const float* wrRow  = wr  + (size_t)n * IN;

    v8f acc = {};
#pragma unroll
    for (int k = 0; k < IN; k += 4) {
        int kb = k + khalf;
        v2f aM; aM.x = aggRow[kb] * invm; aM.y = aggRow[kb + 1] * invm;
        v2f aX; aX.x = xRow[kb];          aX.y = xRow[kb + 1];
        v2f bL; bL.x = wlRow[kb];         bL.y = wlRow[kb + 1];
        v2f bR; bR.x = wrRow[kb];         bR.y = wrRow[kb + 1];
        acc = __builtin_amdgcn_wmma_f32_16x16x4_f32(false, aM, false, bL,
                                                    (short)0, acc, false, false);
        acc = __builtin_amdgcn_wmma_f32_16x16x4_f32(false, aX, false, bR,
                                                    (short)0, acc, false, false);
    }

    const float bcol = bias[n];
    const float a    = PRELU ? slope[0] : 0.0f;
    int rowBase = tr * 16 + hi * 8;
#pragma unroll
    for (int r = 0; r < 8; ++r) {
        float v = acc[r] + bcol;
        if (PRELU) v = fmaxf(v, 0.0f) + a * fminf(v, 0.0f);
        out[(size_t)(rowBase + r) * OUT_DIM + n] = v;
    }
}

// ---------------- final projection: one wave per node -----------------------
__global__ void proj_kernel(const float* __restrict__ h,
                            const float* __restrict__ wp,
                            const float* __restrict__ bp,
                            float* __restrict__ out, int n) {
    int wave = (int)((blockIdx.x * blockDim.x + threadIdx.x) >> 5);
    int lane = threadIdx.x & 31;
    if (wave >= n) return;
    const float2 v = *reinterpret_cast<const float2*>(h  + (size_t)wave * DIM_OUT + lane * 2);
    const float2 w = *reinterpret_cast<const float2*>(wp + lane * 2);
    float s = v.x * w.x + v.y * w.y;
#pragma unroll
    for (int off = 16; off > 0; off >>= 1) s += __shfl_xor(s, off, 32);
    if (lane == 0) out[wave] = s + bp[0];
}

// ---------------------------------------------------------------------------
extern "C" void kernel_launch(void* const* d_in, const int* in_sizes, int n_in,
                              void* d_out, int out_size, void* d_ws, size_t ws_size,
                              hipStream_t stream) {
    (void)in_sizes; (void)n_in; (void)out_size; (void)ws_size;

    const float* x   = (const float*)d_in[0];
    const int*   ei  = (const int*)d_in[1];     // int32 (JAX x64 disabled)
    const float* w1l = (const float*)d_in[2];
    const float* w1r = (const float*)d_in[3];
    const float* b1  = (const float*)d_in[4];
    const float* w2l = (const float*)d_in[5];
    const float* w2r = (const float*)d_in[6];
    const float* b2  = (const float*)d_in[7];
    const float* w3l = (const float*)d_in[8];
    const float* w3r = (const float*)d_in[9];
    const float* b3  = (const float*)d_in[10];
    const float* a   = (const float*)d_in[11];
    const float* wp  = (const float*)d_in[12];
    const float* bp  = (const float*)d_in[13];

    const int* src = ei;
    const int* dst = ei + N_EDGES;

    // workspace layout
    float* agg = (float*)d_ws;                         // N*128
    float* h1  = agg + (size_t)N_NODES * DIM_HID;      // N*128
    float* h2  = h1  + (size_t)N_NODES * DIM_HID;      // N*128
    float* inv = h2  + (size_t)N_NODES * DIM_HID;      // N
    float* h3  = h1;                                   // reuse h1 for N*64

    const int THREADS = 256;
    const int edgeBlocks    = (N_EDGES + THREADS - 1) / THREADS;
    const int nodeBlocks    = (N_NODES + THREADS - 1) / THREADS;
    const int scatterBlocks = (int)(((size_t)N_EDGES * 32 + THREADS - 1) / THREADS);

    // ---- degrees (structure-only, shared by all layers) ----
    hipMemsetAsync(inv, 0, N_NODES * sizeof(float), stream);
    degree_kernel<<<edgeBlocks, THREADS, 0, stream>>>(dst, inv, N_EDGES);
    invcnt_kernel<<<nodeBlocks, THREADS, 0, stream>>>(inv, N_NODES);

    const int tilesHid = (N_NODES / 16) * (DIM_HID / 16);   // 25000
    const int tilesOut = (N_NODES / 16) * (DIM_OUT / 16);   // 12500
    const int gemmHidBlocks = (tilesHid + 7) / 8;           // 8 waves/block
    const int gemmOutBlocks = (tilesOut + 7) / 8;

    // ---- layer 1 ----
    hipMemsetAsync(agg, 0, (size_t)N_NODES * DIM_HID * sizeof(float), stream);
    scatter_kernel<<<scatterBlocks, THREADS, 0, stream>>>(x, src, dst, agg, N_EDGES);
    sage_gemm_kernel<DIM_HID, true><<<gemmHidBlocks, THREADS, 0, stream>>>(
        agg, inv, x, w1l, w1r, b1, a, h1, tilesHid);

    // ---- layer 2 ----
    hipMemsetAsync(agg, 0, (size_t)N_NODES * DIM_HID * sizeof(float), stream);
    scatter_kernel<<<scatterBlocks, THREADS, 0, stream>>>(h1, src, dst, agg, N_EDGES);
    sage_gemm_kernel<DIM_HID, true><<<gemmHidBlocks, THREADS, 0, stream>>>(
        agg, inv, h1, w2l, w2r, b2, a, h2, tilesHid);

    // ---- layer 3 (no activation) ----
    hipMemsetAsync(agg, 0, (size_t)N_NODES * DIM_HID * sizeof(float), stream);
    scatter_kernel<<<scatterBlocks, THREADS, 0, stream>>>(h2, src, dst, agg, N_EDGES);
    sage_gemm_kernel<DIM_OUT, false><<<gemmOutBlocks, THREADS, 0, stream>>>(
        agg, inv, h2, w3l, w3r, b3, a, h3, tilesOut);

    // ---- projection ----
    const int projBlocks = (int)(((size_t)N_NODES * 32 + THREADS - 1) / THREADS);
    proj_kernel<<<projBlocks, THREADS, 0, stream>>>(h3, wp, bp, (float*)d_out, N_NODES);
}